// GIN_2276332667310
// MI455X (gfx1250) — compile-verified
//
#include <hip/hip_runtime.h>

// ---------------- problem constants (match reference) ----------------
#define Nn   100000
#define Ee   1600000
#define DIN  64
#define DH   128
#define DOUT 8
#define GG   512
#define BN_EPS 1e-5f

#define ROW_TILES (Nn / 16)   // 6250, exact

typedef float v2f __attribute__((ext_vector_type(2)));
typedef float v8f __attribute__((ext_vector_type(8)));

// native fp32 global atomic add (global_atomic_add_f32)
__device__ __forceinline__ void atomAddF(float* p, float v) {
#if defined(__HIP_DEVICE_COMPILE__)
  unsafeAtomicAdd(p, v);
#else
  atomicAdd(p, v);
#endif
}

// ---------------- kernels ----------------

// agg[i] = (1+eps) * hin[i]    (compact N x din, same layout in and out)
__global__ void combine_init_k(const float* __restrict__ hin,
                               const float* __restrict__ eps,
                               float* __restrict__ agg, int total) {
  int i = blockIdx.x * blockDim.x + threadIdx.x;
  if (i >= total) return;
  float s = 1.0f + eps[0];
  agg[i] = s * hin[i];
}

// for each edge e: agg[dst[e]][:] += hin[src[e]][:]   (one thread = one float4 chunk;
// for din=128 a full wave covers one edge's 512B row -> coalesced gather)
__global__ void scatter_k(const float* __restrict__ hin,
                          const int* __restrict__ src,
                          const int* __restrict__ dst,
                          float* __restrict__ agg,
                          int din, int total) {
  int t = blockIdx.x * blockDim.x + threadIdx.x;
  if (t >= total) return;
  int chunks = din >> 2;
  int e = t / chunks;
  int c = (t - e * chunks) << 2;
  int s = src[e];
  int d = dst[e];
  float4 v = *reinterpret_cast<const float4*>(hin + (size_t)s * din + c);
  float* q = agg + (size_t)d * din + c;
  atomAddF(q + 0, v.x);
  atomAddF(q + 1, v.y);
  atomAddF(q + 2, v.z);
  atomAddF(q + 3, v.w);
}

// Z[N x 128] = A[N x K] @ W[K x 128] + bias, via V_WMMA_F32_16X16X4_F32.
// Each wave owns a full 16-row x 128-col output stripe: 8 f32 accumulators
// (64 VGPRs of C), so the A fragment is loaded once per K-step and the 8
// WMMAs per step are independent (no D->A/B RAW hazards back-to-back).
// block = 256 threads = 8 waves = 8 row tiles; grid = ceil(6250/8).
__global__ void __launch_bounds__(256) gemm_wmma_k(const float* __restrict__ A, int lda, int K,
                                                   const float* __restrict__ W,
                                                   const float* __restrict__ bias,
                                                   float* __restrict__ Z) {
  const int lane = threadIdx.x & 31;
  const int wave = threadIdx.x >> 5;
  const int tile = blockIdx.x * 8 + wave;     // 16-row tile index
  if (tile >= ROW_TILES) return;              // wave-uniform: EXEC stays all-1s below
  const int row0 = tile * 16;
  const int m    = lane & 15;                 // row (A) / col (B, C, D)
  const int half = lane >> 4;                 // K-pair select for A/B fragments

  v8f acc[8];
#pragma unroll
  for (int c = 0; c < 8; ++c) acc[c] = (v8f){0.f, 0.f, 0.f, 0.f, 0.f, 0.f, 0.f, 0.f};

  const float* arow = A + (size_t)(row0 + m) * lda;

  for (int k = 0; k < K; k += 4) {
    const int kk = k + 2 * half;
    // A 16x4 f32 fragment: VGPR0 = K=kk (lanes->M), VGPR1 = K=kk+1
    v2f a;
    a.x = arow[kk];
    a.y = arow[kk + 1];
    const float* w0 = W + (size_t)kk * DH + m;
#pragma unroll
    for (int c = 0; c < 8; ++c) {
      // B 4x16 f32 fragment for column tile c: VGPR0 = row kk, VGPR1 = row kk+1
      v2f b;
      b.x = w0[c * 16];
      b.y = w0[c * 16 + DH];
      acc[c] = __builtin_amdgcn_wmma_f32_16x16x4_f32(false, a, false, b,
                                                     (short)0, acc[c], false, false);
    }
  }

#pragma unroll
  for (int c = 0; c < 8; ++c) {
    float bv = bias[c * 16 + m];
#pragma unroll
    for (int j = 0; j < 8; ++j) {
      // C/D layout: lane l, VGPR j -> (M = j + 8*half, N = l&15)
      Z[(size_t)(row0 + j + 8 * half) * DH + c * 16 + m] = acc[c][j] + bv;
    }
  }
}

// per-feature sum and sum-of-squares over the node axis -> stats[0..127], stats[128..255]
__global__ void bn_stats_k(const float* __restrict__ Z, float* __restrict__ stats,
                           int rowsPerBlock) {
  int f  = threadIdx.x;                 // 128 threads = 128 features
  int r0 = blockIdx.x * rowsPerBlock;
  int r1 = r0 + rowsPerBlock;
  if (r1 > Nn) r1 = Nn;
  float s = 0.f, q = 0.f;
  for (int r = r0; r < r1; ++r) {
    float v = Z[(size_t)r * DH + f];
    s += v;
    q += v * v;
  }
  atomAddF(&stats[f], s);
  atomAddF(&stats[DH + f], q);
}

// Out = relu(gamma*(Z-mean)*rsqrt(var+eps)+beta), training-mode stats from `stats`
__global__ void bn_apply_relu_k(const float* __restrict__ Z,
                                const float* __restrict__ stats,
                                const float* __restrict__ gamma,
                                const float* __restrict__ beta,
                                float* __restrict__ Out, int total) {
  int i = blockIdx.x * blockDim.x + threadIdx.x;
  if (i >= total) return;
  int f = i & (DH - 1);
  const float inv_n = 1.0f / (float)Nn;
  float mean = stats[f] * inv_n;
  float var  = stats[DH + f] * inv_n - mean * mean;
  float r    = rsqrtf(var + BN_EPS);
  float v    = gamma[f] * (Z[i] - mean) * r + beta[f];
  Out[i] = v > 0.f ? v : 0.f;
}

// pooled[graph_ids[r]][c] += H[r][c]
__global__ void pool_k(const float* __restrict__ H, const int* __restrict__ gid,
                       float* __restrict__ pooled, int din, int total) {
  int i = blockIdx.x * blockDim.x + threadIdx.x;
  if (i >= total) return;
  int r = i / din;
  int c = i - r * din;
  atomAddF(&pooled[(size_t)gid[r] * din + c], H[i]);
}

// score[g][o] += pooled[g] @ W + b
__global__ void head_k(const float* __restrict__ pooled, const float* __restrict__ W,
                       const float* __restrict__ b, float* __restrict__ score, int din) {
  int i = blockIdx.x * blockDim.x + threadIdx.x;
  if (i >= GG * DOUT) return;
  int g = i / DOUT;
  int o = i - g * DOUT;
  float s = b[o];
  const float* p = pooled + (size_t)g * din;
  for (int d = 0; d < din; ++d) s += p[d] * W[d * DOUT + o];
  score[i] += s;
}

// ---------------- host launcher ----------------

static inline unsigned cdiv(long long a, int b) { return (unsigned)((a + b - 1) / b); }

extern "C" void kernel_launch(void* const* d_in, const int* in_sizes, int n_in,
                              void* d_out, int out_size, void* d_ws, size_t ws_size,
                              hipStream_t stream) {
  (void)n_in; (void)out_size; (void)ws_size;

  // --- locate inputs (handle insertion-order vs pytree-sorted flattening) ---
  const float *x; const int *esrc, *edst, *gids;
  int pbase;
  if (in_sizes[0] == Nn * DIN) {
    // top level in setup_inputs() insertion order: x, edge_src, edge_dst, graph_ids, params...
    x    = (const float*)d_in[0];
    esrc = (const int*)d_in[1];
    edst = (const int*)d_in[2];
    gids = (const int*)d_in[3];
    pbase = 4;
  } else {
    // top level sorted: edge_dst, edge_src, graph_ids, params..., x
    edst = (const int*)d_in[0];
    esrc = (const int*)d_in[1];
    gids = (const int*)d_in[2];
    pbase = 3;
    x    = (const float*)d_in[pbase + 44 + 10];
  }

  // per-layer dict: insertion order = eps,W1,b1,g1,be1,W2,b2,g2,be2,gout,bout
  //                 alphabetical    = W1,W2,b1,b2,be1,be2,bout,eps,g1,g2,gout
  bool insertion = (in_sizes[pbase] == 1);
  const float *W1[4], *b1[4], *g1[4], *be1[4], *W2[4], *b2[4], *g2[4], *be2[4];
  const float *gout[4], *bout[4], *epsp[4];
  for (int l = 0; l < 4; ++l) {
    int o = pbase + 11 * l;
    if (insertion) {
      epsp[l] = (const float*)d_in[o + 0];
      W1[l]   = (const float*)d_in[o + 1];  b1[l]  = (const float*)d_in[o + 2];
      g1[l]   = (const float*)d_in[o + 3];  be1[l] = (const float*)d_in[o + 4];
      W2[l]   = (const float*)d_in[o + 5];  b2[l]  = (const float*)d_in[o + 6];
      g2[l]   = (const float*)d_in[o + 7];  be2[l] = (const float*)d_in[o + 8];
      gout[l] = (const float*)d_in[o + 9];  bout[l] = (const float*)d_in[o + 10];
    } else {
      W1[l]   = (const float*)d_in[o + 0];  W2[l]  = (const float*)d_in[o + 1];
      b1[l]   = (const float*)d_in[o + 2];  b2[l]  = (const float*)d_in[o + 3];
      be1[l]  = (const float*)d_in[o + 4];  be2[l] = (const float*)d_in[o + 5];
      bout[l] = (const float*)d_in[o + 6];  epsp[l] = (const float*)d_in[o + 7];
      g1[l]   = (const float*)d_in[o + 8];  g2[l]  = (const float*)d_in[o + 9];
      gout[l] = (const float*)d_in[o + 10];
    }
  }
  const float *predW[5], *predb[5];
  int pb = pbase + 44;
  for (int j = 0; j < 5; ++j) {
    predW[j] = (const float*)d_in[pb + 2 * j + 0];
    predb[j] = (const float*)d_in[pb + 2 * j + 1];
  }

  // --- workspace layout (floats): agg | zbuf | hbuf | stats(256) | pooled(G*128) ---
  float* agg    = (float*)d_ws;
  float* zbuf   = agg  + (size_t)Nn * DH;
  float* hbuf   = zbuf + (size_t)Nn * DH;
  float* stats  = hbuf + (size_t)Nn * DH;
  float* pooled = stats + 2 * DH;
  float* score  = (float*)d_out;

  const unsigned gemmGrid = cdiv(ROW_TILES, 8);

  hipMemsetAsync(score, 0, (size_t)GG * DOUT * sizeof(float), stream);

  // readout head 0 on raw features x
  hipMemsetAsync(pooled, 0, (size_t)GG * DIN * sizeof(float), stream);
  pool_k<<<cdiv((long long)Nn * DIN, 256), 256, 0, stream>>>(x, gids, pooled, DIN, Nn * DIN);
  head_k<<<cdiv(GG * DOUT, 256), 256, 0, stream>>>(pooled, predW[0], predb[0], score, DIN);

  const float* hin = x;
  for (int l = 0; l < 4; ++l) {
    const int din = (l == 0) ? DIN : DH;
    const int totalND = Nn * din;

    // GINConv aggregation: agg = (1+eps)*h + scatter_sum(h[src] -> dst)
    combine_init_k<<<cdiv(totalND, 256), 256, 0, stream>>>(hin, epsp[l], agg, totalND);
    const int totalE = Ee * (din >> 2);
    scatter_k<<<cdiv(totalE, 256), 256, 0, stream>>>(hin, esrc, edst, agg, din, totalE);

    // MLP layer 1: Linear -> BN -> ReLU
    gemm_wmma_k<<<gemmGrid, 256, 0, stream>>>(agg, din, din, W1[l], b1[l], zbuf);
    hipMemsetAsync(stats, 0, 2 * DH * sizeof(float), stream);
    bn_stats_k<<<cdiv(Nn, 256), 128, 0, stream>>>(zbuf, stats, 256);
    bn_apply_relu_k<<<cdiv((long long)Nn * DH, 256), 256, 0, stream>>>(
        zbuf, stats, g1[l], be1[l], agg, Nn * DH);

    // MLP layer 2: Linear -> BN -> ReLU
    gemm_wmma_k<<<gemmGrid, 256, 0, stream>>>(agg, DH, DH, W2[l], b2[l], zbuf);
    hipMemsetAsync(stats, 0, 2 * DH * sizeof(float), stream);
    bn_stats_k<<<cdiv(Nn, 256), 128, 0, stream>>>(zbuf, stats, 256);
    bn_apply_relu_k<<<cdiv((long long)Nn * DH, 256), 256, 0, stream>>>(
        zbuf, stats, g2[l], be2[l], agg, Nn * DH);

    // outer BN -> ReLU
    hipMemsetAsync(stats, 0, 2 * DH * sizeof(float), stream);
    bn_stats_k<<<cdiv(Nn, 256), 128, 0, stream>>>(agg, stats, 256);
    bn_apply_relu_k<<<cdiv((long long)Nn * DH, 256), 256, 0, stream>>>(
        agg, stats, gout[l], bout[l], hbuf, Nn * DH);

    // readout head l+1
    hipMemsetAsync(pooled, 0, (size_t)GG * DH * sizeof(float), stream);
    pool_k<<<cdiv((long long)Nn * DH, 256), 256, 0, stream>>>(hbuf, gids, pooled, DH, Nn * DH);
    head_k<<<cdiv(GG * DOUT, 256), 256, 0, stream>>>(pooled, predW[l + 1], predb[l + 1], score, DH);

    hin = hbuf;
  }
}